// HoloGraph_43774306681402
// MI455X (gfx1250) — compile-verified
//
#include <hip/hip_runtime.h>
#include <hip/hip_bf16.h>

// ---------------- model constants ----------------
#define NNODE 4096
#define FDIM  64
#define CHN   128
#define NOSC  4
#define CGRP  32          // CH / NOSC
#define NCOEF 7
#define GKDIM (FDIM * NCOEF)   // 448
#define QSTEPS 8
#define EPSF  1e-6f

typedef __attribute__((ext_vector_type(2))) float v2f;
typedef __attribute__((ext_vector_type(8))) float v8f;

// ---------------- optional CDNA5 async global->LDS copy ----------------
#ifndef TRY_ASYNC
#define TRY_ASYNC 1
#endif
#if TRY_ASYNC && __has_builtin(__builtin_amdgcn_global_load_async_to_lds_b128)
#define HAVE_ASYNC_LDS 1
typedef int v4i_async __attribute__((vector_size(16)));
typedef __attribute__((address_space(1))) v4i_async glb_v4i;
typedef __attribute__((address_space(3))) v4i_async lds_v4i;
__device__ __forceinline__ void async_cp16(const float* g, float* l) {
    __builtin_amdgcn_global_load_async_to_lds_b128((glb_v4i*)g, (lds_v4i*)l, 0, 0);
}
__device__ __forceinline__ void async_wait0() {
#if __has_builtin(__builtin_amdgcn_s_wait_asynccnt)
    __builtin_amdgcn_s_wait_asynccnt(0);
#else
    asm volatile("s_wait_asynccnt 0x0" ::: "memory");
#endif
}
#else
#define HAVE_ASYNC_LDS 0
#endif

// =====================================================================
// Degree (column sums of adj), deterministic two-stage reduction
// =====================================================================
__global__ void deg_partial(const float* __restrict__ adj, float* __restrict__ part) {
    int col = blockIdx.x * 256 + threadIdx.x;
    int r0  = blockIdx.y * 256;
    float s = 0.f;
    for (int m = r0; m < r0 + 256; ++m) s += adj[(size_t)m * NNODE + col];
    part[(size_t)blockIdx.y * NNODE + col] = s;
}

__global__ void deg_final(const float* __restrict__ part, float* __restrict__ deg) {
    int col = blockIdx.x * 256 + threadIdx.x;
    float s = 0.f;
    for (int r = 0; r < 16; ++r) s += part[(size_t)r * NNODE + col];
    deg[col] = s + EPSF;
}

// v[n,f] = u[n,f] / deg[n]
__global__ void scale_invdeg(const float* __restrict__ u, const float* __restrict__ deg,
                             float* __restrict__ v) {
    int i = blockIdx.x * 256 + threadIdx.x;      // NNODE*FDIM
    v[i] = u[i] / deg[i >> 6];
}

// =====================================================================
// MultiConv1D: 4 branches K=1,3,5,7, node-major output ybr[n, b*64+o]
// =====================================================================
__global__ void multiconv(const float* __restrict__ x,
                          const float* __restrict__ w0, const float* __restrict__ b0,
                          const float* __restrict__ w1, const float* __restrict__ b1,
                          const float* __restrict__ w2, const float* __restrict__ b2,
                          const float* __restrict__ w3, const float* __restrict__ b3,
                          float* __restrict__ ybr) {
    __shared__ float xs[7 * FDIM];
    int n = blockIdx.x;
    int t = threadIdx.x;
    for (int i = t; i < 7 * FDIM; i += 256) {
        int dr = i >> 6, f = i & 63;
        int nn = n + dr - 3;
        xs[i] = (nn >= 0 && nn < NNODE) ? x[(size_t)nn * FDIM + f] : 0.f;
    }
    __syncthreads();
    int br = t >> 6;            // branch
    int o  = t & 63;            // out channel within branch
    const float* w; const float* bb; int K;
    switch (br) {
        case 0:  w = w0; bb = b0; K = 1; break;
        case 1:  w = w1; bb = b1; K = 3; break;
        case 2:  w = w2; bb = b2; K = 5; break;
        default: w = w3; bb = b3; K = 7; break;
    }
    int pad = K >> 1;
    float s = bb[o];
    for (int k = 0; k < K; ++k) {
        int roff = (3 + k - pad) * FDIM;   // window row in LDS
        for (int ci = 0; ci < FDIM; ++ci)
            s += w[((size_t)o * FDIM + ci) * K + k] * xs[roff + ci];
    }
    ybr[(size_t)n * 256 + t] = s;
}

// =====================================================================
// LDS-tiled WMMA f32 16x16x4 GEMM: dst = adj[4096,4096] @ B[4096, Ncols]
// PMODE=1 -> dst = 0.5*(src + adj@B)
// Block = 256 threads = 8 waves; output tile 32 rows x 64 cols; KT = 64.
// =====================================================================
#define KT   64
#define APAD 68
#define BPAD 68

template <int PMODE>
__global__ __launch_bounds__(256)
void adj_gemm(const float* __restrict__ adj, const float* __restrict__ B, int ldb,
              const float* __restrict__ src, float* __restrict__ dst, int Ncols) {
    __shared__ float aT[32 * APAD];
    __shared__ float bT[KT * BPAD];
    int t = threadIdx.x;
    int wave = t >> 5, lane = t & 31;
    int mtsub = wave >> 2;              // 0..1
    int ntsub = wave & 3;               // 0..3
    int mbase = blockIdx.x * 32;
    int nbase = blockIdx.y * 64;
    int mrow = lane & 15;
    int ncol = lane & 15;
    int koff = (lane >> 4) << 1;        // 0 or 2
    v8f acc = {};

    for (int k0 = 0; k0 < NNODE; k0 += KT) {
        __syncthreads();                 // WAR: previous tile fully consumed
#if HAVE_ASYNC_LDS
#pragma unroll
        for (int i = 0; i < 2; ++i) {    // adj tile: 32x64 = 512 float4
            int id = t + i * 256;
            int r = id >> 4, c4 = (id & 15) << 2;
            async_cp16(adj + (size_t)(mbase + r) * NNODE + k0 + c4, aT + r * APAD + c4);
        }
#pragma unroll
        for (int i = 0; i < 4; ++i) {    // B tile: 64x64 = 1024 float4
            int id = t + i * 256;
            int r = id >> 4, c4 = (id & 15) << 2;
            async_cp16(B + (size_t)(k0 + r) * ldb + nbase + c4, bT + r * BPAD + c4);
        }
        async_wait0();
#else
#pragma unroll
        for (int i = 0; i < 2; ++i) {
            int id = t + i * 256;
            int r = id >> 4, c4 = (id & 15) << 2;
            *(float4*)(aT + r * APAD + c4) =
                *(const float4*)(adj + (size_t)(mbase + r) * NNODE + k0 + c4);
        }
#pragma unroll
        for (int i = 0; i < 4; ++i) {
            int id = t + i * 256;
            int r = id >> 4, c4 = (id & 15) << 2;
            *(float4*)(bT + r * BPAD + c4) =
                *(const float4*)(B + (size_t)(k0 + r) * ldb + nbase + c4);
        }
#endif
        __syncthreads();
#pragma unroll
        for (int k2 = 0; k2 < KT; k2 += 4) {
            v2f a, b;
            const float* ap = aT + (mtsub * 16 + mrow) * APAD + k2 + koff;
            a.x = ap[0];
            a.y = ap[1];
            const float* bp = bT + (k2 + koff) * BPAD + ntsub * 16 + ncol;
            b.x = bp[0];
            b.y = bp[BPAD];
            acc = __builtin_amdgcn_wmma_f32_16x16x4_f32(false, a, false, b,
                                                        (short)0, acc, false, false);
        }
    }
    int m0 = mbase + mtsub * 16;
    int n0 = nbase + ntsub * 16;
    int radd = (lane >> 4) << 3;
#pragma unroll
    for (int r = 0; r < 8; ++r) {
        int row = m0 + r + radd;
        int col = n0 + ncol;
        float vv = acc[r];
        if (PMODE) vv = 0.5f * (src[(size_t)row * Ncols + col] + vv);
        dst[(size_t)row * Ncols + col] = vv;
    }
}

// =====================================================================
// WMMA f32 GEMM: dst[m,n] = sum_k A[m,k]*W[n,k] + bias[n] (+ extra[m,n])
// W [Ncols,K] row-major, staged (zero-padded) in LDS.
// Block = 256 threads = 8 waves = 8 M-tiles x 1 N-tile (16 cols).
// Dynamic LDS: 16*(K+4) floats.
// =====================================================================
__global__ __launch_bounds__(256)
void gemm_wt(const float* __restrict__ A, int lda,
             const float* __restrict__ W,
             const float* __restrict__ bias,
             const float* __restrict__ extra, int lde,
             float* __restrict__ dst, int ldd,
             int Ncols, int K) {
    extern __shared__ float wT[];          // [16][K+4]
    __shared__ float bls[16];
    const int Kpad = K + 4;
    int t = threadIdx.x;
    int wave = t >> 5, lane = t & 31;
    int mt = blockIdx.x * 8 + wave;
    int nt = blockIdx.y;
    int m0 = mt * 16, n0 = nt * 16;

    // stage W tile (rows = output cols), zero-padded for invalid cols
    int nq = K >> 2;                       // float4s per row
    for (int id = t; id < 16 * nq; id += 256) {
        int r  = id / nq;
        int c4 = (id - r * nq) << 2;
        int nW = n0 + r;
        float4 wv = make_float4(0.f, 0.f, 0.f, 0.f);
        if (nW < Ncols) wv = *(const float4*)(W + (size_t)nW * K + c4);
        *(float4*)(wT + r * Kpad + c4) = wv;
    }
    if (t < 16) {
        int nW = n0 + t;
        bls[t] = (nW < Ncols) ? bias[nW] : 0.f;
    }
    __syncthreads();

    int mrow = lane & 15;
    int ncol = lane & 15;
    int koff = (lane >> 4) << 1;
    v8f acc = {};
    const float* arow = A + (size_t)(m0 + mrow) * lda;
    const float* wrow = wT + ncol * Kpad;
    for (int k = 0; k < K; k += 4) {
        v2f a, b;
        a.x = arow[k + koff];
        a.y = arow[k + koff + 1];
        b.x = wrow[k + koff];
        b.y = wrow[k + koff + 1];
        acc = __builtin_amdgcn_wmma_f32_16x16x4_f32(false, a, false, b,
                                                    (short)0, acc, false, false);
    }
    float bs = bls[ncol];
    int radd = (lane >> 4) << 3;
#pragma unroll
    for (int r = 0; r < 8; ++r) {
        int row = m0 + r + radd;
        int col = n0 + ncol;
        if (col < Ncols) {
            float vv = acc[r] + bs;
            if (extra) vv += extra[(size_t)row * lde + col];
            dst[(size_t)row * ldd + col] = vv;
        }
    }
}

// =====================================================================
// GST helpers
// =====================================================================
__global__ void diff_abs(const float* __restrict__ a, const float* __restrict__ b,
                         float* __restrict__ dstA, float* __restrict__ dstG) {
    int i = blockIdx.x * 256 + threadIdx.x;     // NNODE*FDIM
    float d = fabsf(a[i] - b[i]);
    if (dstA) dstA[i] = d;
    if (dstG) {
        int n = i >> 6, f = i & 63;
        dstG[(size_t)n * GKDIM + f * NCOEF] = d;
    }
}

__global__ void copy_coef0(const float* __restrict__ x, float* __restrict__ G) {
    int i = blockIdx.x * 256 + threadIdx.x;     // NNODE*FDIM
    int n = i >> 6, f = i & 63;
    G[(size_t)n * GKDIM + f * NCOEF] = x[i];
}

// =====================================================================
// Kuramoto elementwise kernels (node-major state S[n, g*4+o])
// =====================================================================
__global__ void unit_norm(float* __restrict__ S) {
    int i = blockIdx.x * 256 + threadIdx.x;     // NNODE*CGRP
    int n = i >> 5, g = i & 31;
    float* p = S + (size_t)n * CHN + g * 4;
    float a = p[0], b = p[1], c = p[2], d = p[3];
    float nr = sqrtf(a * a + b * b + c * c + d * d) + EPSF;
    p[0] = a / nr; p[1] = b / nr; p[2] = c / nr; p[3] = d / nr;
}

__global__ void gn_reduce(const float* __restrict__ cmat, float* __restrict__ stat) {
    __shared__ float ssum[256], ssq[256];
    int g = blockIdx.x;
    float s = 0.f, q = 0.f;
    for (int t = threadIdx.x; t < NNODE * NOSC; t += 256) {
        int n = t >> 2, o = t & 3;
        float v = cmat[(size_t)n * CHN + g * 4 + o];
        s += v; q += v * v;
    }
    ssum[threadIdx.x] = s; ssq[threadIdx.x] = q;
    __syncthreads();
    for (int st = 128; st > 0; st >>= 1) {
        if (threadIdx.x < st) {
            ssum[threadIdx.x] += ssum[threadIdx.x + st];
            ssq[threadIdx.x]  += ssq[threadIdx.x + st];
        }
        __syncthreads();
    }
    if (threadIdx.x == 0) {
        const float inv = 1.0f / (float)(NNODE * NOSC);
        float mu = ssum[0] * inv;
        float var = ssq[0] * inv - mu * mu;
        stat[g] = mu;
        stat[32 + g] = rsqrtf(var + 1e-5f);
    }
}

__global__ void kur_update(const float* __restrict__ cmat, const float* __restrict__ stat,
                           const float* __restrict__ gn_g, const float* __restrict__ gn_b,
                           float* __restrict__ S) {
    int i = blockIdx.x * 256 + threadIdx.x;     // NNODE*CGRP
    int n = i >> 5, g = i & 31;
    float mu = stat[g], rstd = stat[32 + g];
    const float* cp = cmat + (size_t)n * CHN + g * 4;
    float* xp = S + (size_t)n * CHN + g * 4;
    float cn[4], xv[4];
#pragma unroll
    for (int o = 0; o < 4; ++o) {
        cn[o] = (cp[o] - mu) * rstd * gn_g[g * 4 + o] + gn_b[g * 4 + o];
        xv[o] = xp[o];
    }
    float inner = cn[0] * xv[0] + cn[1] * xv[1] + cn[2] * xv[2] + cn[3] * xv[3];
    float xn[4];
    float nrm2 = 0.f;
#pragma unroll
    for (int o = 0; o < 4; ++o) {
        xn[o] = xv[o] + (cn[o] - inner * xv[o]);   // GAMMA = 1
        nrm2 += xn[o] * xn[o];
    }
    float nr = sqrtf(nrm2) + EPSF;
#pragma unroll
    for (int o = 0; o < 4; ++o) xp[o] = xn[o] / nr;
}

__global__ void rotor_mag(const float* __restrict__ v, float* __restrict__ mag) {
    int i = blockIdx.x * 256 + threadIdx.x;     // NNODE*CGRP
    int n = i >> 5, g = i & 31;
    const float* p = v + (size_t)n * CHN + g * 4;
    mag[(size_t)n * CGRP + g] =
        sqrtf(p[0] * p[0] + p[1] * p[1] + p[2] * p[2] + p[3] * p[3] + EPSF);
}

// =====================================================================
// Host orchestration
// =====================================================================
extern "C" void kernel_launch(void* const* d_in, const int* in_sizes, int n_in,
                              void* d_out, int out_size, void* d_ws, size_t ws_size,
                              hipStream_t stream) {
    const float* x      = (const float*)d_in[0];
    const float* adj    = (const float*)d_in[1];
    const float* mc_w[4] = {(const float*)d_in[2], (const float*)d_in[4],
                            (const float*)d_in[6], (const float*)d_in[8]};
    const float* mc_b[4] = {(const float*)d_in[3], (const float*)d_in[5],
                            (const float*)d_in[7], (const float*)d_in[9]};
    const float* proj_w = (const float*)d_in[10];
    const float* proj_b = (const float*)d_in[11];
    const float* pat_w  = (const float*)d_in[12];
    const float* pat_b  = (const float*)d_in[13];
    const float* kur_wc = (const float*)d_in[14];
    const float* kur_bc = (const float*)d_in[15];
    const float* kur_wy = (const float*)d_in[16];
    const float* kur_by = (const float*)d_in[17];
    const float* gn_g   = (const float*)d_in[18];
    const float* gn_b   = (const float*)d_in[19];
    const float* ro_w1  = (const float*)d_in[20];
    const float* ro_b1  = (const float*)d_in[21];
    const float* ro_w2  = (const float*)d_in[22];
    const float* ro_b2  = (const float*)d_in[23];
    const float* out_w  = (const float*)d_in[24];
    const float* out_b  = (const float*)d_in[25];
    float* out = (float*)d_out;

    // ---- workspace layout (float offsets) ----
    const size_t NF = (size_t)NNODE * FDIM;          // 262144
    const size_t NC = (size_t)NNODE * CHN;           // 524288
    float* ws   = (float*)d_ws;
    float* deg  = ws;                                 // 4096
    float* v    = deg + NNODE;                        // NF
    float* bu1  = v   + NF;                           // NF
    float* bu2  = bu1 + NF;                           // NF
    float* bu3  = bu2 + NF;                           // NF
    float* s10  = bu3 + NF;                           // NF
    float* s11  = s10 + NF;                           // NF
    float* G    = s11 + NF;                           // NNODE*448
    float* cy   = G + (size_t)NNODE * GKDIM;          // NC
    float* S    = cy + NC;                            // NC
    float* agg  = S + NC;                             // NC
    float* cmat = agg + NC;                           // NC
    float* stat = cmat + NC;                          // 64
    // phase-based aliases (dead regions reused)
    float* ybr  = G;                                  // [N,256] (before GST)
    float* ymat = G + (size_t)NNODE * 256;            // [N,128] (before GST)
    float* part = bu1;                                // deg partials (before GST)
    float* vro  = agg;                                // after Kuramoto
    float* mag  = bu1;                                // after GST
    float* xs   = bu2;                                // after GST
    (void)in_sizes; (void)n_in; (void)out_size; (void)ws_size;

    const dim3 B256(256);
    auto launch_wt = [&](const float* A, int lda, const float* W, const float* bias,
                         const float* extra, int lde, float* dst, int ldd,
                         int Ncols, int K) {
        size_t sh = (size_t)16 * (K + 4) * sizeof(float);
        gemm_wt<<<dim3(32, (Ncols + 15) / 16), B256, sh, stream>>>(
            A, lda, W, bias, extra, lde, dst, ldd, Ncols, K);
    };

    // ---- degrees ----
    deg_partial<<<dim3(16, 16), B256, 0, stream>>>(adj, part);
    deg_final<<<dim3(16), B256, 0, stream>>>(part, deg);

    // ---- control pattern y, cy ----
    multiconv<<<dim3(NNODE), B256, 0, stream>>>(x, mc_w[0], mc_b[0], mc_w[1], mc_b[1],
                                                mc_w[2], mc_b[2], mc_w[3], mc_b[3], ybr);
    launch_wt(ybr, 256, proj_w, proj_b, nullptr, 0, ymat, CHN, CHN, 256);
    launch_wt(ymat, CHN, kur_wy, kur_by, nullptr, 0, cy, CHN, CHN, CHN);

    // ---- GST ----
    auto P = [&](const float* src, float* dst) {
        scale_invdeg<<<dim3(NF / 256), B256, 0, stream>>>(src, deg, v);
        adj_gemm<1><<<dim3(NNODE / 32, FDIM / 64), B256, 0, stream>>>(adj, v, FDIM,
                                                                      src, dst, FDIM);
    };
    const dim3 GD(NF / 256);
    copy_coef0<<<GD, B256, 0, stream>>>(x, G);
    // wavelets(x): coefs 1..3 (+ s1_0, s1_1 buffers)
    P(x, bu1); P(bu1, bu2);
    diff_abs<<<GD, B256, 0, stream>>>(x, bu1, s10, G + 1);       // |psi0|
    diff_abs<<<GD, B256, 0, stream>>>(bu1, bu2, s11, G + 2);     // |psi1|
    P(bu2, bu3); P(bu3, bu1);                                    // u3, u4
    diff_abs<<<GD, B256, 0, stream>>>(bu2, bu1, nullptr, G + 3); // |psi2|
    // wavelets(s1_0): coefs 4,5
    P(s10, bu1); P(bu1, bu2);
    diff_abs<<<GD, B256, 0, stream>>>(bu1, bu2, nullptr, G + 4);
    P(bu2, bu3); P(bu3, bu1);
    diff_abs<<<GD, B256, 0, stream>>>(bu2, bu1, nullptr, G + 5);
    // wavelets(s1_1): coef 6 (psi2 only)
    P(s11, bu1); P(bu1, bu2);
    P(bu2, bu3); P(bu3, bu1);
    diff_abs<<<GD, B256, 0, stream>>>(bu2, bu1, nullptr, G + 6);

    // ---- x0 = G @ pat_w^T + pat_b ; unit-norm groups ----
    launch_wt(G, GKDIM, pat_w, pat_b, nullptr, 0, S, CHN, CHN, GKDIM);
    unit_norm<<<dim3((NNODE * CGRP) / 256), B256, 0, stream>>>(S);

    // ---- Kuramoto dynamics ----
    for (int q = 0; q < QSTEPS; ++q) {
        adj_gemm<0><<<dim3(NNODE / 32, CHN / 64), B256, 0, stream>>>(adj, S, CHN,
                                                                     nullptr, agg, CHN);
        launch_wt(agg, CHN, kur_wc, kur_bc, cy, CHN, cmat, CHN, CHN, CHN);
        gn_reduce<<<dim3(CGRP), B256, 0, stream>>>(cmat, stat);
        kur_update<<<dim3((NNODE * CGRP) / 256), B256, 0, stream>>>(cmat, stat,
                                                                    gn_g, gn_b, S);
    }

    // ---- readout ----
    launch_wt(S, CHN, ro_w1, ro_b1, nullptr, 0, vro, CHN, CHN, CHN);
    rotor_mag<<<dim3((NNODE * CGRP) / 256), B256, 0, stream>>>(vro, mag);
    launch_wt(mag, CGRP, ro_w2, ro_b2, nullptr, 0, xs, CHN, CHN, CGRP);
    launch_wt(xs, CHN, out_w, out_b, nullptr, 0, out, 4, 4, CHN);
}